// OSFCNFwSTU_80264348828099
// MI455X (gfx1250) — compile-verified
//
#include <hip/hip_runtime.h>
#include <math.h>

// ---------------------------------------------------------------------------
// Online spectral-filtering controller scan (T=256 sequential steps) for
// MI455X / gfx1250.  Persistent 32-WG kernel, device-wide barriers between
// phases, fp32 WMMA (v_wmma_f32_16x16x4_f32) for the v = W@phi GEMM, and
// async global->LDS staging (global_load_async_to_lds_b128) for the vhat
// buffers consumed by the fused M-update pass.
// ---------------------------------------------------------------------------

typedef __attribute__((ext_vector_type(2))) float v2f;
typedef __attribute__((ext_vector_type(8))) float v8f;

#define NWG    32
#define NTHR   256
#define TSTEPS 256
#define ETA_F  1e-3f

// sizes
#define DD  256   // state dim
#define NN  128   // control dim
#define HH  32    // filters
#define HS  20    // stu filters
#define MH  512   // history length (phi rows)

// workspace layout (float offsets)
static constexpr size_t WS_M   = 0;                        // 32*128*256 = 1048576
static constexpr size_t WS_VH0 = 1048576;                  // 8192 (vhat, parity 0)
static constexpr size_t WS_VH1 = WS_VH0 + 8192;            // 8192 (vhat, parity 1)
static constexpr size_t WS_WHZ = WS_VH1 + 8192;            // 3*256 zero pad rows
static constexpr size_t WS_WH  = WS_WHZ + 768;             // 257*256 w-history
static constexpr size_t WS_UH  = WS_WH + 257 * 256;        // 256*128 u-history (time-major)
static constexpr size_t WS_UHN = WS_UH + 256 * 128;        // 128*256 u-history (n-major)
static constexpr size_t WS_X0  = WS_UHN + 128 * 256;       // 256 (x, parity 0)
static constexpr size_t WS_X1  = WS_X0 + 256;              // 256 (x, parity 1)
static constexpr size_t WS_G   = WS_X1 + 256;              // 128 g = 2*R*u
static constexpr size_t WS_KX  = WS_G + 128;               // 128 K@x
static constexpr size_t WS_QP  = WS_KX + 128;              // 128 xQx partials
static constexpr size_t WS_CN  = WS_QP + 128;              // 256 cn partials (2 halves)
static constexpr size_t WS_Y   = WS_CN + 256;              // 20*128 Y = U2@phi_stu
static constexpr size_t WS_BAR = WS_Y + 2560;              // 2 x u32 barrier state

__device__ __forceinline__ float wred(float v) {
#pragma unroll
  for (int o = 16; o > 0; o >>= 1) v += __shfl_down(v, o, 32);
  return v;
}

__device__ __forceinline__ void grid_sync(unsigned* cnt, unsigned* gen) {
  __syncthreads();
  if (threadIdx.x == 0) {
    __threadfence();
    unsigned g = __hip_atomic_load(gen, __ATOMIC_RELAXED, __HIP_MEMORY_SCOPE_AGENT);
    if (__hip_atomic_fetch_add(cnt, 1u, __ATOMIC_ACQ_REL, __HIP_MEMORY_SCOPE_AGENT) ==
        (unsigned)(NWG - 1)) {
      __hip_atomic_store(cnt, 0u, __ATOMIC_RELAXED, __HIP_MEMORY_SCOPE_AGENT);
      __hip_atomic_fetch_add(gen, 1u, __ATOMIC_ACQ_REL, __HIP_MEMORY_SCOPE_AGENT);
    } else {
      while (__hip_atomic_load(gen, __ATOMIC_ACQUIRE, __HIP_MEMORY_SCOPE_AGENT) == g) {
        __builtin_amdgcn_s_sleep(2);
      }
    }
    __threadfence();
  }
  __syncthreads();
}

// one async 16B global->LDS copy per call (ASYNCcnt-tracked)
__device__ __forceinline__ void async_copy_b128(unsigned lds_off, const float* gaddr) {
  asm volatile("global_load_async_to_lds_b128 %0, %1, off"
               :
               : "v"(lds_off), "v"(gaddr)
               : "memory");
}

__global__ void osfc_init_kernel(const float* __restrict__ x0,
                                 const float* __restrict__ W0,
                                 float* __restrict__ ws) {
  const int tid = threadIdx.x;
  if (tid < DD) {
    ws[WS_X0 + tid] = x0[tid];            // x at parity 0 (t = 0 reads parity 0)
    ws[WS_WH + tid] = W0[tid * MH];       // wh[0] = W0 column 0
  }
  for (int i = tid; i < 768; i += NTHR) ws[WS_WHZ + i] = 0.0f;  // zero pad rows
  if (tid < NN) ws[WS_G + tid] = 0.0f;    // g_{-1} = 0
  if (tid == 0) {
    unsigned* b = (unsigned*)(ws + WS_BAR);
    b[0] = 0u;
    b[1] = 0u;
  }
}

__global__ __launch_bounds__(NTHR) void osfc_scan_kernel(
    const float* __restrict__ Amat, const float* __restrict__ Bmat,
    const float* __restrict__ Qmat, const float* __restrict__ Rmat,
    const float* __restrict__ Kmat, const float* __restrict__ phi,
    const float* __restrict__ sigma, const float* __restrict__ phistu,
    const float* __restrict__ Mstu, float* __restrict__ ws,
    float* __restrict__ out) {
  float* M    = ws + WS_M;
  float* wh   = ws + WS_WH;   // rows -3..-1 are a zero pad (WS_WHZ)
  float* uh   = ws + WS_UH;
  float* uhn  = ws + WS_UHN;
  float* gbuf = ws + WS_G;
  float* kx   = ws + WS_KX;
  float* qp   = ws + WS_QP;
  float* cnp  = ws + WS_CN;
  float* Yb   = ws + WS_Y;
  unsigned* bar = (unsigned*)(ws + WS_BAR);

  const int tid  = threadIdx.x;
  const int bid  = blockIdx.x;
  const int lane = tid & 31;
  const int wid  = tid >> 5;
  const int wv   = bid * 8 + wid;   // 0..255 global wave id

  // Phase-A tile owned by this WG: 16x16 tile of v (d0.., h0..)
  const int d0 = (bid & 15) * 16;
  const int h0 = (bid >> 4) * 16;

  __shared__ float sred[8 * 32 * 8];  // phase-A WMMA partials (8 KB)
  __shared__ float sVhP[16 * 256];    // phase-B vhat_prev slice (16 KB)
  __shared__ float sVhC[16 * 256];    // phase-B vhat_cur  slice (16 KB)
  __shared__ float su[NN];            // phase-C u
  __shared__ float s1[NN], s2[NN];    // phase-C reductions

  // constants hoisted out of the time loop
  const int dA   = d0 + (lane & 15);
  const int hB   = h0 + (lane & 15);
  const int jofs = (lane >> 4) << 1;          // lanes 16-31 hold K=2,3 of chunk
  const float s4r = sqrtf(sqrtf(sigma[hB]));  // sigma^0.25 for this lane's h
  const int nB   = wv & 127;                  // phase-B row
  const int half = wv >> 7;                   // phase-B h-half (uniform per WG)

  for (int t = 0; t < TSTEPS; ++t) {
    float* vhC = ws + ((t & 1) ? WS_VH1 : WS_VH0);
    float* vhP = ws + ((t & 1) ? WS_VH0 : WS_VH1);
    const float* xcur = ws + ((t & 1) ? WS_X1 : WS_X0);
    float* xnxt = ws + ((t & 1) ? WS_X0 : WS_X1);

    // ======== Phase A: v[d,h] = sum_{j<=t} wh[t-j][d] * phi[j][h] (WMMA f32) ==
    // Branchless: wh rows -3..-1 are zeros, so lanes with j>t read 0.
    {
      v8f c = {0.f, 0.f, 0.f, 0.f, 0.f, 0.f, 0.f, 0.f};
      for (int jb = 4 * wid; jb <= t; jb += 32) {  // 8 waves split K, uniform trip
        const int j0 = jb + jofs, j1 = j0 + 1;
        v2f a, b;
        a.x = wh[(t - j0) * DD + dA];
        a.y = wh[(t - j1) * DD + dA];
        b.x = phi[j0 * HH + hB];
        b.y = phi[j1 * HH + hB];
        c = __builtin_amdgcn_wmma_f32_16x16x4_f32(false, a, false, b, (short)0, c,
                                                  false, false);
      }
#pragma unroll
      for (int r = 0; r < 8; ++r) sred[(wid * 32 + lane) * 8 + r] = c[r];
    }
    __syncthreads();
    if (wid == 0) {  // combine K-partials (fixed order), scale by sigma^0.25
#pragma unroll
      for (int r = 0; r < 8; ++r) {
        float tot = 0.0f;
#pragma unroll
        for (int w = 0; w < 8; ++w) tot += sred[(w * 32 + lane) * 8 + r];
        const int dd = d0 + r + ((lane >> 4) << 3);  // C layout: M=r / r+8
        vhC[hB * DD + dd] = s4r * tot;               // vhat = sig4 .* v
      }
    }
    grid_sync(bar + 0, bar + 1);

    // ======== Phase B: fused M update (prev grad) + cn partials; Kx/Qx ========
    {
      // Async-stage the 16 h-rows of vhat_prev/cur this WG consumes (8x reuse).
      const float* gP = vhP + half * 16 * DD;
      const float* gC = vhC + half * 16 * DD;
#pragma unroll
      for (int r = 0; r < 4; ++r) {
        const int fo = (tid + NTHR * r) * 4;  // float offset of this 16B chunk
        async_copy_b128((unsigned)(uintptr_t)&sVhP[fo], gP + fo);
        async_copy_b128((unsigned)(uintptr_t)&sVhC[fo], gC + fo);
      }
      asm volatile("s_wait_asynccnt 0" ::: "memory");
      __syncthreads();

      const float e = ETA_F * gbuf[nB];  // ETA * g_{t-1}[n]
      float acc = 0.0f;
      for (int hl = 0; hl < 16; ++hl) {
        const int bm = ((half * 16 + hl) * NN + nB) * DD;
        if (hl + 1 < 16)
          __builtin_prefetch(&M[bm + NN * DD + lane * 4], 1, 1);
#pragma unroll
        for (int cc = 0; cc < 2; ++cc) {
          const int off = cc * 128 + lane * 4;
          float4 m  = *(const float4*)(M + bm + off);
          float4 vp = *(const float4*)(sVhP + hl * DD + off);
          float4 vc = *(const float4*)(sVhC + hl * DD + off);
          m.x -= e * vp.x; m.y -= e * vp.y; m.z -= e * vp.z; m.w -= e * vp.w;
          *(float4*)(M + bm + off) = m;
          acc += m.x * vc.x + m.y * vc.y + m.z * vc.z + m.w * vc.w;
        }
      }
      acc = wred(acc);
      if (lane == 0) cnp[half * NN + nB] = acc;

      if (half == 0) {  // Kx[n]
        float p = 0.0f;
        for (int d = lane; d < DD; d += 32) p += Kmat[nB * DD + d] * xcur[d];
        p = wred(p);
        if (lane == 0) kx[nB] = p;
      } else {          // xQx partials: rows 2n, 2n+1
        float p = 0.0f;
#pragma unroll
        for (int rr = 0; rr < 2; ++rr) {
          const int r = nB * 2 + rr;
          float q = 0.0f;
          for (int d = lane; d < DD; d += 32) q += Qmat[r * DD + d] * xcur[d];
          q = wred(q);
          if (lane == 0) p += xcur[r] * q;
        }
        if (lane == 0) qp[nB] = p;
      }
    }
    grid_sync(bar + 0, bar + 1);

    // ======== Phase C (WG0): u, g = 2Ru, cost -> out[t] =======================
    if (bid == 0) {
      if (tid < NN) {
        const float uu = cnp[tid] + cnp[NN + tid] - kx[tid];  // u = -Kx + cn
        su[tid] = uu;
        uh[t * NN + tid]  = uu;   // time-major (phase E)
        uhn[tid * 256 + t] = uu;  // n-major (phase D, coalesced)
      }
      __syncthreads();
      float part = 0.0f, part2 = 0.0f;
      if (tid < NN) {
        float ru = 0.0f;
        for (int m2 = 0; m2 < NN; ++m2)
          ru += Rmat[m2 * NN + tid] * su[m2];  // R symmetric -> coalesced
        gbuf[tid] = 2.0f * ru;        // grad seed for next step's M update
        part  = su[tid] * ru;         // u^T R u partial
        part2 = qp[tid];              // x^T Q x partial
      }
      if (tid < NN) { s1[tid] = part; s2[tid] = part2; }
      __syncthreads();
      for (int st = 64; st > 0; st >>= 1) {
        if (tid < st) { s1[tid] += s1[tid + st]; s2[tid] += s2[tid + st]; }
        __syncthreads();
      }
      if (tid == 0) out[t] = s1[0] + s2[0];
    }
    grid_sync(bar + 0, bar + 1);

    // ======== Phase D: Y[n,k] = sum_{j<=t} uh[t-j][n] * phi_stu[j][k] =========
    if (wv < NN) {
      const int n = wv;
      float acc[HS];
#pragma unroll
      for (int k = 0; k < HS; ++k) acc[k] = 0.0f;
      for (int j = lane; j <= t; j += 32) {
        const float uv = uhn[n * 256 + (t - j)];  // coalesced across lanes
        const float* pr = phistu + j * HS;
#pragma unroll
        for (int k = 0; k < HS; ++k) acc[k] += uv * pr[k];
      }
#pragma unroll
      for (int k = 0; k < HS; ++k) {
        const float v = wred(acc[k]);
        if (lane == 0) Yb[k * NN + n] = v;
      }
    }
    grid_sync(bar + 0, bar + 1);

    // ======== Phase E: x_next, new_w (one wave per d) ==========================
    {
      const int d = wv;  // 0..255
      float xn = 0.0f, ab = 0.0f;
#pragma unroll
      for (int k = 0; k < HS; ++k) {
        const float* mr = Mstu + (size_t)(k * DD + d) * NN;
        const float* yr = Yb + k * NN;
        for (int n2 = lane; n2 < NN; n2 += 32) xn += mr[n2] * yr[n2];
      }
      for (int c2 = lane; c2 < DD; c2 += 32) ab += Amat[d * DD + c2] * xcur[c2];
      const float* ut = uh + t * NN;
      for (int n2 = lane; n2 < NN; n2 += 32) ab += Bmat[d * NN + n2] * ut[n2];
      xn = wred(xn);
      ab = wred(ab);
      if (lane == 0) {
        xnxt[d] = xn;                      // x_{t+1}
        wh[(t + 1) * DD + d] = xn - ab;    // new_w history entry
      }
    }
    grid_sync(bar + 0, bar + 1);
  }
}

extern "C" void kernel_launch(void* const* d_in, const int* in_sizes, int n_in,
                              void* d_out, int out_size, void* d_ws, size_t ws_size,
                              hipStream_t stream) {
  (void)in_sizes; (void)n_in; (void)out_size; (void)ws_size;
  const float* Amat   = (const float*)d_in[0];
  const float* Bmat   = (const float*)d_in[1];
  const float* Qmat   = (const float*)d_in[2];
  const float* Rmat   = (const float*)d_in[3];
  const float* Kmat   = (const float*)d_in[4];
  const float* phi    = (const float*)d_in[5];
  const float* sigma  = (const float*)d_in[6];
  const float* phistu = (const float*)d_in[7];
  const float* M0     = (const float*)d_in[8];
  const float* Mstu   = (const float*)d_in[9];
  const float* x0     = (const float*)d_in[10];
  const float* W0     = (const float*)d_in[11];
  float* ws  = (float*)d_ws;
  float* out = (float*)d_out;

  // M working copy (updated in place each step) — re-seeded every launch.
  hipMemcpyAsync(ws + WS_M, M0, (size_t)HH * NN * DD * sizeof(float),
                 hipMemcpyDeviceToDevice, stream);
  osfc_init_kernel<<<1, NTHR, 0, stream>>>(x0, W0, ws);
  osfc_scan_kernel<<<NWG, NTHR, 0, stream>>>(Amat, Bmat, Qmat, Rmat, Kmat, phi,
                                             sigma, phistu, Mstu, ws, out);
}